// DynamicSphericalHarmonicsPrior_68616397521241
// MI455X (gfx1250) — compile-verified
//
#include <hip/hip_runtime.h>
#include <hip/hip_bf16.h>
#include <math.h>

// Problem constants (from reference):
//  x: (1, L=4, C=128, H=240, W=480) fp32 ; Wa: (128,64) ; ba: (64)
//  Wb: (64, 1312) ; bb: (1312) ; gain: (128) ; Y: (36, 115200)
#define LDIM 4
#define CDIM 128
#define CH   64          // C/2
#define HWD  115200      // 240*480
#define KDIM 36          // LMAX^2
#define RDIM 8
#define WFD  1312        // C*R + R*K = 1024 + 288
#define NTILES (HWD / 16)       // 7200 column tiles of 16
#define NT_PER_BLOCK 8          // N tiles per block strip (7200 % 8 == 0)

typedef __attribute__((ext_vector_type(2))) float v2f;
typedef __attribute__((ext_vector_type(8))) float v8f;

// ---------------------------------------------------------------------------
// Kernel 1: global mean pool  ctx[l][c] = mean_p x[l][c][p]
// ---------------------------------------------------------------------------
__global__ __launch_bounds__(256) void dshp_pool_kernel(
    const float* __restrict__ x, float* __restrict__ ctx) {
  const int c = blockIdx.x;
  const int l = blockIdx.y;
  const float* p = x + ((size_t)(l * CDIM + c)) * HWD;
  float s = 0.0f;
  for (int i = threadIdx.x; i < HWD; i += 256) s += p[i];
  __shared__ float red[256];
  red[threadIdx.x] = s;
  __syncthreads();
  if (threadIdx.x < 128) red[threadIdx.x] += red[threadIdx.x + 128];
  __syncthreads();
  if (threadIdx.x < 64) red[threadIdx.x] += red[threadIdx.x + 64];
  __syncthreads();
  if (threadIdx.x < 32) {
    float v = red[threadIdx.x] + red[threadIdx.x + 32];
    // wave32 shuffle reduction
    for (int off = 16; off > 0; off >>= 1) v += __shfl_down(v, off, 32);
    if (threadIdx.x == 0) ctx[l * CDIM + c] = v * (1.0f / (float)HWD);
  }
}

// ---------------------------------------------------------------------------
// Kernel 2: hypernet (tiny). One block per l.
//   h  = silu(ctx @ Wa + ba)            (64)
//   wf = h @ Wb + bb                    (1312)
//   coeff[c][k] = sum_r wf[c*8+r] * wf[1024 + r*36 + k]
// ---------------------------------------------------------------------------
__global__ __launch_bounds__(256) void dshp_hyper_kernel(
    const float* __restrict__ ctx, const float* __restrict__ Wa,
    const float* __restrict__ ba,  const float* __restrict__ Wb,
    const float* __restrict__ bb,  float* __restrict__ coeff) {
  const int l = blockIdx.x;
  const int t = threadIdx.x;
  __shared__ float sctx[CDIM];
  __shared__ float sh[CH];
  __shared__ float swf[WFD];

  if (t < CDIM) sctx[t] = ctx[l * CDIM + t];
  __syncthreads();

  if (t < CH) {
    float acc = ba[t];
    for (int c = 0; c < CDIM; ++c) acc += sctx[c] * Wa[c * CH + t];
    sh[t] = acc / (1.0f + __expf(-acc));   // silu
  }
  __syncthreads();

  for (int o = t; o < WFD; o += 256) {
    float acc = bb[o];
    for (int j = 0; j < CH; ++j) acc += sh[j] * Wb[j * WFD + o];
    swf[o] = acc;
  }
  __syncthreads();

  for (int ck = t; ck < CDIM * KDIM; ck += 256) {
    const int c = ck / KDIM;
    const int k = ck - c * KDIM;
    float acc = 0.0f;
    #pragma unroll
    for (int r = 0; r < RDIM; ++r)
      acc += swf[c * RDIM + r] * swf[CDIM * RDIM + r * KDIM + k];
    coeff[((size_t)l * CDIM + c) * KDIM + k] = acc;
  }
}

// ---------------------------------------------------------------------------
// Kernel 3: out[l,c,p] = x[l,c,p] + gain[c] * sum_k coeff[l,c,k] * Y[k,p]
// WMMA f32 16x16x4, K=36 -> 9 k-steps. One wave = one 16-row M tile of one l.
// Block = 8 waves = all 8 M tiles of one l; covers NT_PER_BLOCK N tiles.
//
// VGPR layouts (ISA 7.12.2):
//  A 16x4 f32:  lanes 0-15 -> M=lane, regs {K=0,K=1}; lanes 16-31 -> {K=2,K=3}
//  B 4x16 f32:  reg0 = {K=0 | K=2 by lane half}, reg1 = {K=1 | K=3}, N = lane&15
//  C/D 16x16:   reg r -> row r + 8*(lane>=16), col = lane&15
// ---------------------------------------------------------------------------
__global__ __launch_bounds__(256) void dshp_bias_wmma_kernel(
    const float* __restrict__ x, const float* __restrict__ coeff,
    const float* __restrict__ Y, const float* __restrict__ gain,
    float* __restrict__ out) {
  const int lane  = threadIdx.x & 31;
  const int wave  = threadIdx.x >> 5;        // M tile index 0..7
  const int l     = blockIdx.y;
  const int m0    = wave * 16;
  const int col   = lane & 15;
  const int khalf = (lane >> 4) * 2;         // 0 for lanes 0-15, 2 for 16-31
  const int roff  = (lane >> 4) * 8;         // C/D row offset per lane half

  // ---- A fragments: coeff[l][m0 + (lane&15)][ks*4 + khalf + {0,1}] ----
  const float* cf = coeff + ((size_t)l * CDIM + (m0 + col)) * KDIM;
  v2f afrag[9];
  #pragma unroll
  for (int ks = 0; ks < 9; ++ks) {
    afrag[ks].x = cf[ks * 4 + khalf + 0];
    afrag[ks].y = cf[ks * 4 + khalf + 1];
  }

  // per-lane gains for the 8 accumulator rows
  float g[8];
  #pragma unroll
  for (int r = 0; r < 8; ++r) g[r] = gain[m0 + roff + r];

  const size_t rowbase = ((size_t)l * CDIM + (m0 + roff)) * HWD;

  for (int t = 0; t < NT_PER_BLOCK; ++t) {
    const int n0 = (blockIdx.x * NT_PER_BLOCK + t) * 16;
    const int n  = n0 + col;

    // prefetch next tile's x rows (gfx1250 global_prefetch path)
    if (t + 1 < NT_PER_BLOCK)
      __builtin_prefetch(x + rowbase + (size_t)n + 16, 0, 0);

    v8f acc = {};
    #pragma unroll
    for (int ks = 0; ks < 9; ++ks) {
      const int kb = ks * 4 + khalf;
      v2f bfrag;
      bfrag.x = Y[(size_t)(kb + 0) * HWD + n];
      bfrag.y = Y[(size_t)(kb + 1) * HWD + n];
      acc = __builtin_amdgcn_wmma_f32_16x16x4_f32(
          /*neg_a=*/false, afrag[ks], /*neg_b=*/false, bfrag,
          /*c_mod=*/(short)0, acc, /*reuse_a=*/false, /*reuse_b=*/false);
    }

    // fused epilogue: out = x + gain * acc
    const float* xp = x + rowbase + (size_t)n;
    float*       op = out + rowbase + (size_t)n;
    #pragma unroll
    for (int r = 0; r < 8; ++r) {
      op[(size_t)r * HWD] = xp[(size_t)r * HWD] + g[r] * acc[r];
    }
  }
}

// ---------------------------------------------------------------------------
extern "C" void kernel_launch(void* const* d_in, const int* in_sizes, int n_in,
                              void* d_out, int out_size, void* d_ws, size_t ws_size,
                              hipStream_t stream) {
  const float* x    = (const float*)d_in[0];   // (1,4,128,240,480)
  const float* Wa   = (const float*)d_in[1];   // (128,64)
  const float* ba   = (const float*)d_in[2];   // (64)
  const float* Wb   = (const float*)d_in[3];   // (64,1312)
  const float* bb   = (const float*)d_in[4];   // (1312)
  const float* gain = (const float*)d_in[5];   // (128)
  const float* Y    = (const float*)d_in[6];   // (36,115200)
  float* out = (float*)d_out;

  // workspace: ctx (L*C floats) then coeff (L*C*K floats)
  float* ctx   = (float*)d_ws;
  float* coeff = ctx + (LDIM * CDIM);

  dshp_pool_kernel<<<dim3(CDIM, LDIM), 256, 0, stream>>>(x, ctx);
  dshp_hyper_kernel<<<dim3(LDIM), 256, 0, stream>>>(ctx, Wa, ba, Wb, bb, coeff);
  dshp_bias_wmma_kernel<<<dim3(NTILES / NT_PER_BLOCK, LDIM), 256, 0, stream>>>(
      x, coeff, Y, gain, out);
}